// BasePolicyNetwork_86088324481362
// MI455X (gfx1250) — compile-verified
//
#include <hip/hip_runtime.h>
#include <cstdint>

// ---------------------------------------------------------------------------
// out = normalize(gamma(softmax(state @ wv^T + prior)))   (q,k paths are dead)
// state = [bias | weight]  -> fold bias*wv[:,0] into C init, GEMM K = 512.
// Data movement: weight streamed HBM -> LDS via GLOBAL_LOAD_ASYNC_TO_LDS_B128
// (no staging VGPRs, no spills), compute via V_WMMA_F32_16X16X4_F32.
// ---------------------------------------------------------------------------

typedef __attribute__((ext_vector_type(2))) float v2f;
typedef __attribute__((ext_vector_type(8))) float v8f;

#define N_ROWS   131072
#define ROWS_BLK 128        // rows per workgroup (8 waves * 16)
#define KDIM     512        // GEMM K after folding bias column
#define KC       64         // K chunk staged in LDS
#define ASTR     68         // LDS row stride (floats) for A tile (272 B, 16B-aligned)
#define BSTR     516        // LDS row stride (floats) for B matrix
#define OUTC     20

// --- stateless RNG helpers --------------------------------------------------
__device__ __forceinline__ uint32_t pcg_hash(uint32_t x) {
    x = x * 747796405u + 2891336453u;
    uint32_t w = ((x >> ((x >> 28u) + 4u)) ^ x) * 277803737u;
    return (w >> 22u) ^ w;
}
__device__ __forceinline__ float u01(uint32_t& s) {
    s = pcg_hash(s);
    return ((float)(s >> 8) + 0.5f) * (1.0f / 16777216.0f);
}
__device__ __forceinline__ float bm_normal(uint32_t& s) {
    float u1 = u01(s), u2 = u01(s);
    return sqrtf(-2.0f * __logf(u1)) * __cosf(6.28318530718f * u2);
}
// Marsaglia-Tsang gamma(alpha,1) with alpha<1 boost; bounded rejection.
__device__ __forceinline__ float gamma_sample(float alpha, uint32_t& s) {
    float boost = 1.0f;
    if (alpha < 1.0f) {
        float u = u01(s);
        boost = __powf(u, 1.0f / fmaxf(alpha, 1e-8f));
        alpha += 1.0f;
    }
    float d = alpha - (1.0f / 3.0f);
    float c = rsqrtf(9.0f * d);
    #pragma unroll 1
    for (int it = 0; it < 8; ++it) {
        float x = bm_normal(s);
        float t = 1.0f + c * x;
        if (t <= 0.0f) continue;
        float v = t * t * t;
        float u = u01(s);
        if (__logf(u) < 0.5f * x * x + d - d * v + d * __logf(v))
            return d * v * boost;
    }
    return d * boost;
}

__global__ __launch_bounds__(256) void
dirichlet_head_kernel(const float* __restrict__ bias,
                      const float* __restrict__ weight,   // [N, 512] row-major
                      const float* __restrict__ prior,    // [N, 20]
                      const float* __restrict__ wv,       // [20, 513] row-major
                      float* __restrict__ out)            // [N, 20]
{
    __shared__ float sA[2][ROWS_BLK * ASTR];   // 2 x 34,816 B  (double buffer)
    __shared__ float sB[32 * BSTR];            // 66,048 B      (wv, padded to 32 ch)
    __shared__ float sEpi[8][16 * 33];         // 16,896 B      (per-wave C spill)

    const int t    = threadIdx.x;
    const int wave = t >> 5;
    const int lane = t & 31;
    const int r0   = blockIdx.x * ROWS_BLK;

    const int mrow  = wave * 16 + (lane & 15);       // A-row this lane feeds
    const int koff  = (lane < 16) ? 0 : 2;           // K offset per half-wave
    const int half8 = (lane < 16) ? 0 : 8;           // C-layout M offset
    const int n0    = lane & 15;                     // B/C column, tile 0
    const int n1    = 16 + (lane & 15);              // B/C column, tile 1

    // Per-thread async-copy geometry: 8 x b128 per chunk, fully coalesced.
    const int srow = t >> 4;                         // staging base row (+i*16)
    const int sc4  = (t & 15) * 4;                   // staging column (floats)
    // LDS byte addresses (generic addr truncated to 32 bits == LDS offset)
    const uint32_t ldsA0 = (uint32_t)(uintptr_t)(&sA[0][0]);
    const uint32_t ldsA1 = (uint32_t)(uintptr_t)(&sA[1][0]);

    // ---- C init: fold bias * wv[:,0] -------------------------------------
    const float wv0_0 = wv[n0 * 513];                         // n0 < 20 always
    const float wv0_1 = (n1 < OUTC) ? wv[n1 * 513] : 0.0f;
    v8f acc0, acc1;
    #pragma unroll
    for (int i = 0; i < 8; ++i) {
        float bz = bias[r0 + wave * 16 + i + half8];
        acc0[i] = bz * wv0_0;
        acc1[i] = bz * wv0_1;
    }

    // ---- issue async copy of A chunk 0 -----------------------------------
    #pragma unroll
    for (int i = 0; i < 8; ++i) {
        int row = srow + i * 16;
        uint32_t ldst = ldsA0 + (uint32_t)(row * ASTR + sc4) * 4u;
        uint32_t goff = ((uint32_t)(r0 + row) * KDIM + 0 + sc4) * 4u;
        asm volatile("global_load_async_to_lds_b128 %0, %1, %2"
                     :: "v"(ldst), "v"(goff), "s"(weight) : "memory");
    }

    // ---- stage B = wv[:,1:513] zero-padded to 32 channels (plain loads) ---
    #pragma unroll 4
    for (int i = 0; i < 64; ++i) {               // 32*512 / 256 threads
        int flat = t + i * 256;
        int n = flat >> 9;
        int k = flat & 511;
        sB[n * BSTR + k] = (n < OUTC) ? wv[n * 513 + k + 1] : 0.0f;
    }

    asm volatile("s_wait_asynccnt 0x0" ::: "memory");
    __syncthreads();

    // ---- main K loop: async double-buffer + FP32 WMMA --------------------
    #pragma unroll 1
    for (int c = 0; c < KDIM / KC; ++c) {
        const int buf = c & 1;
        const int kc  = c * KC;

        if (c < (KDIM / KC) - 1) {
            // issue async copy of chunk c+1 into the other buffer
            const uint32_t ldsNext = (buf == 0) ? ldsA1 : ldsA0;
            #pragma unroll
            for (int i = 0; i < 8; ++i) {
                int row = srow + i * 16;
                uint32_t ldst = ldsNext + (uint32_t)(row * ASTR + sc4) * 4u;
                uint32_t goff = ((uint32_t)(r0 + row) * KDIM + kc + KC + sc4) * 4u;
                asm volatile("global_load_async_to_lds_b128 %0, %1, %2"
                             :: "v"(ldst), "v"(goff), "s"(weight) : "memory");
            }
        }

        const float* aBase  = &sA[buf][mrow * ASTR + koff];
        const float* bBase0 = &sB[n0 * BSTR + kc + koff];
        const float* bBase1 = &sB[n1 * BSTR + kc + koff];
        #pragma unroll
        for (int s = 0; s < KC / 4; ++s) {
            v2f a  = *(const v2f*)(aBase  + s * 4);
            v2f b0 = *(const v2f*)(bBase0 + s * 4);
            v2f b1 = *(const v2f*)(bBase1 + s * 4);
            acc0 = __builtin_amdgcn_wmma_f32_16x16x4_f32(
                       false, a, false, b0, (short)0, acc0, false, false);
            acc1 = __builtin_amdgcn_wmma_f32_16x16x4_f32(
                       false, a, false, b1, (short)0, acc1, false, false);
        }

        // own async writes landed; then block-wide visibility
        asm volatile("s_wait_asynccnt 0x0" ::: "memory");
        __syncthreads();
    }

    // ---- spill C tiles to LDS so each lane owns a full row ---------------
    #pragma unroll
    for (int i = 0; i < 8; ++i) {
        sEpi[wave][(i + half8) * 33 + n0]      = acc0[i];
        sEpi[wave][(i + half8) * 33 + 16 + n0] = acc1[i];
    }
    __syncthreads();

    // ---- epilogue: softmax -> gamma -> normalize (16 rows per wave) ------
    if (lane < 16) {
        const int r = r0 + wave * 16 + lane;
        float v[OUTC];
        float mx = -1e30f;
        #pragma unroll
        for (int j = 0; j < OUTC; ++j) {
            v[j] = sEpi[wave][lane * 33 + j] + prior[(size_t)r * OUTC + j];
            mx = fmaxf(mx, v[j]);
        }
        float ssum = 0.0f;
        #pragma unroll
        for (int j = 0; j < OUTC; ++j) {
            v[j] = __expf(v[j] - mx);
            ssum += v[j];
        }
        const float inv = 1.0f / ssum;

        float g[OUTC];
        float gsum = 0.0f;
        #pragma unroll 1
        for (int j = 0; j < OUTC; ++j) {
            uint32_t seed = pcg_hash((uint32_t)r * 20u + (uint32_t)j + 0x9E3779B9u);
            g[j] = gamma_sample(v[j] * inv, seed);
            gsum += g[j];
        }
        const float ginv = 1.0f / fmaxf(gsum, 1e-30f);
        #pragma unroll
        for (int j = 0; j < OUTC; ++j)
            out[(size_t)r * OUTC + j] = g[j] * ginv;
    }
}

extern "C" void kernel_launch(void* const* d_in, const int* in_sizes, int n_in,
                              void* d_out, int out_size, void* d_ws, size_t ws_size,
                              hipStream_t stream) {
    (void)in_sizes; (void)n_in; (void)out_size; (void)d_ws; (void)ws_size;
    const float* bias   = (const float*)d_in[0];
    const float* weight = (const float*)d_in[1];
    const float* prior  = (const float*)d_in[2];
    // d_in[3]=wq, d_in[4]=wk dead (attention window size 1); d_in[6..7] rel_* dead
    const float* wv     = (const float*)d_in[5];
    float* out          = (float*)d_out;

    dim3 grid(N_ROWS / ROWS_BLK);   // 1024
    dim3 block(256);                // 8 wave32
    hipLaunchKernelGGL(dirichlet_head_kernel, grid, block, 0, stream,
                       bias, weight, prior, wv, out);
}